// MutualInformationLoss_14276471292473
// MI455X (gfx1250) — compile-verified
//
#include <hip/hip_runtime.h>
#include <hip/hip_bf16.h>

#define NUM_BINS 64
#define N_IMG    4
#define NPIX     (512 * 512)          // per image
#define TOT      (N_IMG * NPIX)       // 1048576
#define MI_EPS   1.1920929e-7f        // float32 eps
#define INV_2SIG2 204800.0f           // 1 / (2 * (0.1/64)^2)
#define HSCALE   (1.0f / (float)TOT)
#define WS_HIST_OFF 16                // floats; ws[0..3] = min1,max1,min2,max2

typedef float v2f __attribute__((ext_vector_type(2)));
typedef float v8f __attribute__((ext_vector_type(8)));

// ---------------------------------------------------------------- init ----
__global__ void mi_init_kernel(float* ws) {
    int t = threadIdx.x;                       // blockDim.x == 256
    ws[WS_HIST_OFF + t] = 0.0f;                // zero hist[4][64]
    if (t == 0) {
        ((int*)ws)[0] = 0x7F800000;            // +inf  (min img1)
        ((int*)ws)[1] = (int)0xFF800000;       // -inf  (max img1)
        ((int*)ws)[2] = 0x7F800000;            // +inf  (min img2)
        ((int*)ws)[3] = (int)0xFF800000;       // -inf  (max img2)
    }
}

// ------------------------------------------------------------- min/max ----
__device__ __forceinline__ void atomicMinF(float* a, float v) {
    if (v >= 0.0f) atomicMin((int*)a, __float_as_int(v));
    else           atomicMax((unsigned int*)a, (unsigned int)__float_as_int(v));
}
__device__ __forceinline__ void atomicMaxF(float* a, float v) {
    if (v >= 0.0f) atomicMax((int*)a, __float_as_int(v));
    else           atomicMin((unsigned int*)a, (unsigned int)__float_as_int(v));
}

__global__ void mi_minmax_kernel(const float* __restrict__ img1,
                                 const float* __restrict__ img2,
                                 float* ws) {
    int tid    = blockIdx.x * blockDim.x + threadIdx.x;
    int stride = gridDim.x * blockDim.x;
    float mn1 = INFINITY, mx1 = -INFINITY, mn2 = INFINITY, mx2 = -INFINITY;
    for (int i = tid; i < TOT; i += stride) {
        float a = img1[i], b = img2[i];
        mn1 = fminf(mn1, a); mx1 = fmaxf(mx1, a);
        mn2 = fminf(mn2, b); mx2 = fmaxf(mx2, b);
    }
    // wave32 butterfly reduction
    #pragma unroll
    for (int off = 16; off >= 1; off >>= 1) {
        mn1 = fminf(mn1, __shfl_xor(mn1, off, 32));
        mx1 = fmaxf(mx1, __shfl_xor(mx1, off, 32));
        mn2 = fminf(mn2, __shfl_xor(mn2, off, 32));
        mx2 = fmaxf(mx2, __shfl_xor(mx2, off, 32));
    }
    __shared__ float s[4][8];
    int wave = threadIdx.x >> 5, lane = threadIdx.x & 31;
    if (lane == 0) { s[0][wave] = mn1; s[1][wave] = mx1; s[2][wave] = mn2; s[3][wave] = mx2; }
    __syncthreads();
    if (threadIdx.x == 0) {
        float a = s[0][0], b = s[1][0], c = s[2][0], d = s[3][0];
        int nw = blockDim.x >> 5;
        for (int w = 1; w < nw; ++w) {
            a = fminf(a, s[0][w]); b = fmaxf(b, s[1][w]);
            c = fminf(c, s[2][w]); d = fmaxf(d, s[3][w]);
        }
        atomicMinF(&ws[0], a); atomicMaxF(&ws[1], b);
        atomicMinF(&ws[2], c); atomicMaxF(&ws[3], d);
    }
}

// ----------------------------------------------------------- histogram ----
// grid = (256, N_IMG), block = 256. Lane l owns bins l and l+32; pixel
// values are broadcast across the wave via __shfl so accumulation stays in
// registers (1 exp per pixel-bin pair = the minimal transcendental count).
__global__ void mi_hist_kernel(const float* __restrict__ img1,
                               const float* __restrict__ img2,
                               float* ws) {
    const int n    = blockIdx.y;
    const int lane = threadIdx.x & 31;
    const int wave = threadIdx.x >> 5;

    const float mn1 = ws[0], mx1 = ws[1], mn2 = ws[2], mx2 = ws[3];
    const float inv1 = 1.0f / (mx1 - mn1 + MI_EPS);
    const float inv2 = 1.0f / (mx2 - mn2 + MI_EPS);

    const float c0 = ((float)lane        + 0.5f) * (1.0f / NUM_BINS);
    const float c1 = ((float)(lane + 32) + 0.5f) * (1.0f / NUM_BINS);

    const float* p1 = img1 + (size_t)n * NPIX;
    const float* p2 = img2 + (size_t)n * NPIX;

    float h0 = 0.0f, h1 = 0.0f;
    const int stride = gridDim.x * blockDim.x;            // 65536; NPIX/stride = 4 exactly
    for (int i = blockIdx.x * blockDim.x + threadIdx.x; i < NPIX; i += stride) {
        if (i + stride < NPIX) {                          // gfx1250 global_prefetch_b8
            __builtin_prefetch(p1 + i + stride, 0, 3);
            __builtin_prefetch(p2 + i + stride, 0, 3);
        }
        const float x = (p1[i] - mn1) * inv1;
        const float y = (p2[i] - mn2) * inv2;
        #pragma unroll 8
        for (int j = 0; j < 32; ++j) {
            const float xj = __shfl(x, j, 32);
            const float yj = __shfl(y, j, 32);
            const float dx0 = xj - c0, dy0 = yj - c0;
            const float dx1 = xj - c1, dy1 = yj - c1;
            h0 += __expf(-(dx0 * dx0 + dy0 * dy0) * INV_2SIG2);
            h1 += __expf(-(dx1 * dx1 + dy1 * dy1) * INV_2SIG2);
        }
    }

    __shared__ float lds[8 * NUM_BINS];
    lds[wave * NUM_BINS + lane]      = h0;
    lds[wave * NUM_BINS + lane + 32] = h1;
    __syncthreads();
    const int nw = blockDim.x >> 5;
    if ((int)threadIdx.x < NUM_BINS) {
        float s = 0.0f;
        for (int w = 0; w < nw; ++w) s += lds[w * NUM_BINS + threadIdx.x];
        atomicAdd(&ws[WS_HIST_OFF + n * NUM_BINS + threadIdx.x], s);
    }
}

// ------------------------------------------------------------ finalize ----
// Single wave32 block. Row sums s_n = sum_b hist[n][b] computed with an
// accumulating V_WMMA_F32_16X16X4_F32 chain (B = ones, 16 chunks of K=4).
// A layout per ISA 7.12.2: lanes 0-15 -> M=lane (K0 in v[0], K1 in v[1]);
// lanes 16-31 -> M=lane-16 (K2, K3). C[m,0] lands on lane 0, VGPR m.
__global__ void mi_final_kernel(const float* __restrict__ ws, float* __restrict__ out) {
    const int lane = threadIdx.x;                 // 0..31, exactly one wave
    const float* hist = ws + WS_HIST_OFF;

    const int   m       = lane & 15;
    const int   kb      = (lane >> 4) * 2;        // 0 or 2
    const float rowmask = (m < N_IMG) ? 1.0f : 0.0f;
    const float* rowp   = hist + (m & 3) * NUM_BINS;

    v2f b; b.x = 1.0f; b.y = 1.0f;                // B = ones => C[m,*] = row sum
    v8f c = {};
    #pragma unroll
    for (int t = 0; t < 16; ++t) {
        v2f a;
        a.x = rowp[4 * t + kb    ] * rowmask;
        a.y = rowp[4 * t + kb + 1] * rowmask;
        c = __builtin_amdgcn_wmma_f32_16x16x4_f32(false, a, false, b,
                                                  (short)0, c, false, false);
    }
    __builtin_amdgcn_s_wait_tensorcnt(0);

    // broadcast raw row sums from lane 0 (element (m, N=0) lives there)
    float S[N_IMG];
    S[0] = __shfl(c[0], 0, 32) * HSCALE;
    S[1] = __shfl(c[1], 0, 32) * HSCALE;
    S[2] = __shfl(c[2], 0, 32) * HSCALE;
    S[3] = __shfl(c[3], 0, 32) * HSCALE;

    float inv[N_IMG], px[N_IMG];
    #pragma unroll
    for (int n = 0; n < N_IMG; ++n) {
        inv[n] = 1.0f / (S[n] + MI_EPS);
        px[n]  = S[n] * inv[n];                   // sum_b pxy[n,b]
    }

    // lane owns bins b0 = lane, b1 = lane + 32
    float pxy0[N_IMG], pxy1[N_IMG], py0 = 0.0f, py1 = 0.0f;
    #pragma unroll
    for (int n = 0; n < N_IMG; ++n) {
        pxy0[n] = hist[n * NUM_BINS + lane]      * HSCALE * inv[n];
        pxy1[n] = hist[n * NUM_BINS + lane + 32] * HSCALE * inv[n];
        py0 += pxy0[n];
        py1 += pxy1[n];
    }
    float acc = 0.0f;
    #pragma unroll
    for (int n = 0; n < N_IMG; ++n) {
        acc += pxy0[n] * __logf((pxy0[n] + MI_EPS) / (px[n] * py0 + MI_EPS) + MI_EPS);
        acc += pxy1[n] * __logf((pxy1[n] + MI_EPS) / (px[n] * py1 + MI_EPS) + MI_EPS);
    }
    #pragma unroll
    for (int off = 16; off >= 1; off >>= 1)
        acc += __shfl_xor(acc, off, 32);
    if (lane == 0) out[0] = acc;
}

// -------------------------------------------------------------- launch ----
extern "C" void kernel_launch(void* const* d_in, const int* in_sizes, int n_in,
                              void* d_out, int out_size, void* d_ws, size_t ws_size,
                              hipStream_t stream) {
    const float* img1 = (const float*)d_in[0];
    const float* img2 = (const float*)d_in[1];
    float* out = (float*)d_out;
    float* ws  = (float*)d_ws;   // needs (16 + 256) floats

    mi_init_kernel<<<1, 256, 0, stream>>>(ws);
    mi_minmax_kernel<<<512, 256, 0, stream>>>(img1, img2, ws);
    mi_hist_kernel<<<dim3(256, N_IMG), 256, 0, stream>>>(img1, img2, ws);
    mi_final_kernel<<<1, 32, 0, stream>>>(ws, out);
}